// HierarchicalTimeAttention_30872224923943
// MI455X (gfx1250) — compile-verified
//
#include <hip/hip_runtime.h>
#include <hip/hip_bf16.h>
#include <stdint.h>

#define NN 50000
#define NE 600000
#define DD 128
#define NCLUST 8
#define CN (NCLUST * NN)                      // 400000 segments
#define CND ((size_t)CN * (size_t)DD)         // 51,200,000 floats

static constexpr float SCALE_V = 0.0883883476483184405f;   // 128^-0.5

typedef float v2f __attribute__((ext_vector_type(2)));
typedef float v4f __attribute__((ext_vector_type(4)));
typedef float v8f __attribute__((ext_vector_type(8)));

__device__ __forceinline__ v8f wmma_f32(v2f a, v2f b, v8f c) {
    // D = A(16x4,f32) * B(4x16,f32) + C(16x16,f32)
    return __builtin_amdgcn_wmma_f32_16x16x4_f32(false, a, false, b, (short)0, c,
                                                 false, false);
}

// monotone float<->uint order encoding (so atomicMax(uint) == float max)
__device__ __forceinline__ unsigned ordEnc(float f) {
    unsigned u = __float_as_uint(f);
    return (u & 0x80000000u) ? ~u : (u | 0x80000000u);
}
__device__ __forceinline__ float ordDec(unsigned u) {
    unsigned b = (u & 0x80000000u) ? (u ^ 0x80000000u) : ~u;
    return __uint_as_float(b);
}
#define ENC_NEG_INF 0x007FFFFFu   // ordEnc(-inf)

// ---------------------------------------------------------------------------
// Swizzled LDS image of a 128x128 row-major weight matrix.
//   element W[k][col]  ->  wlds[(k>>1)*256 + col'*2 + (k&1)]
//   col' = (col + 16*((k>>1)&1)) & 127
// So the WMMA B operand (W[K][col], W[K+1][col]) for even K is one aligned
// b64 load, and the two wave-halves (kp vs kp+1) hit disjoint LDS bank sets.
// ---------------------------------------------------------------------------
__device__ __forceinline__ void load_w_lds_swz(float* wlds, const float* __restrict__ W,
                                               int tid) {
    const v4f* s = (const v4f*)W;
#pragma unroll
    for (int i = 0; i < 16; ++i) {
        int vidx = tid + i * 256;            // v4f index into W
        v4f val = s[vidx];
        int k = vidx >> 5;                   // row (vidx*4 / 128)
        int col0 = (vidx & 31) * 4;
        int kp = k >> 1;
        int m = k & 1;
        int rot = (kp & 1) << 4;
#pragma unroll
        for (int e = 0; e < 4; ++e) {
            int colp = (col0 + e + rot) & 127;
            wlds[kp * 256 + colp * 2 + m] = val[e];
        }
    }
}

// B operand fetch from the swizzled image (K = kb + 2*half, col = nb*16+l15)
__device__ __forceinline__ v2f wmma_b_ld(const float* wlds, int kb, int half, int colp) {
    return *(const v2f*)&wlds[((kb >> 1) + half) * 256 + colp * 2];
}

// ---------------------------------------------------------------- init
__global__ void init_kernel(float* __restrict__ sums, float* __restrict__ denom,
                            int* __restrict__ counts, unsigned* __restrict__ segmax,
                            int* __restrict__ epcflag) {
    size_t i = (size_t)blockIdx.x * 256 + threadIdx.x;
    if (i < CND) sums[i] = 0.0f;
    if (i < (size_t)CN) {
        denom[i] = 0.0f;
        counts[i] = 0;
        segmax[i] = ENC_NEG_INF;
    }
    if (i < NCLUST) epcflag[i] = 0;
}

// ---------------------------------------------------------------- q = X*Wq + bq
__global__ __launch_bounds__(256) void qgemm_kernel(const float* __restrict__ X,
                                                    const float* __restrict__ W,
                                                    const float* __restrict__ bias,
                                                    float* __restrict__ Y, int nrows) {
    __shared__ float wlds[DD * DD];
    int tid = threadIdx.x;
    load_w_lds_swz(wlds, W, tid);
    __syncthreads();
    int wave = tid >> 5, lane = tid & 31;
    int half = lane >> 4, l15 = lane & 15, koff = half * 2;
    int cbase = l15 + (half << 4);
    int rowBase = blockIdx.x * 128 + wave * 16;
    int rc = (rowBase + l15 < nrows) ? (rowBase + l15) : 0;
    const float* xrow = X + (size_t)rc * DD;

    v8f acc[8] = {};
    v2f a = *(const v2f*)(xrow + koff);                 // pipelined A load
    for (int kb = 0; kb < DD; kb += 4) {
        v2f an = *(const v2f*)(xrow + ((kb + 4) & (DD - 1)) + koff);
#pragma unroll
        for (int nb = 0; nb < 8; ++nb) {
            int colp = (nb * 16 + cbase) & 127;
            acc[nb] = wmma_f32(a, wmma_b_ld(wlds, kb, half, colp), acc[nb]);
        }
        a = an;
    }
#pragma unroll
    for (int nb = 0; nb < 8; ++nb) {
        int col = nb * 16 + l15;
        float bb = bias[col];
#pragma unroll
        for (int j = 0; j < 8; ++j) {
            int r = rowBase + half * 8 + j;
            if (r < nrows) Y[(size_t)r * DD + col] = acc[nb][j] + bb;
        }
    }
}

// ------------------------------------------- pass B: k, cluster argmax, attn, seg stats
__global__ __launch_bounds__(256) void passB_kernel(
    const float* __restrict__ TF, const float* __restrict__ Wk, const float* __restrict__ bk,
    const float* __restrict__ CE, const float* __restrict__ q, const int* __restrict__ srcIdx,
    float* __restrict__ attnOut, int* __restrict__ segOut,
    int* __restrict__ counts, unsigned* __restrict__ segmax) {
    __shared__ float wlds[DD * DD];
    int tid = threadIdx.x;
    load_w_lds_swz(wlds, Wk, tid);
    __syncthreads();
    int wave = tid >> 5, lane = tid & 31;
    int half = lane >> 4, l15 = lane & 15, koff = half * 2;
    int cbase = l15 + (half << 4);
    int eBase = blockIdx.x * 128 + wave * 16;
    int ec = (eBase + l15 < NE) ? (eBase + l15) : 0;
    const float* trow = TF + (size_t)ec * DD;
    const float* cerow = CE + (size_t)(l15 & 7) * DD;
    int mysrc = srcIdx[ec];   // src of edge (eBase + l15)

    v8f acc[8] = {};
    v8f sacc = {};
    v2f a = *(const v2f*)(trow + koff);                 // pipelined A load
    for (int kb = 0; kb < DD; kb += 4) {
        v2f an = *(const v2f*)(trow + ((kb + 4) & (DD - 1)) + koff);
        v2f bc = *(const v2f*)(cerow + kb + koff);      // B = cluster_emb^T (dup'd cols)
        sacc = wmma_f32(a, bc, sacc);
#pragma unroll
        for (int nb = 0; nb < 8; ++nb) {
            int colp = (nb * 16 + cbase) & 127;
            acc[nb] = wmma_f32(a, wmma_b_ld(wlds, kb, half, colp), acc[nb]);
        }
        a = an;
    }
    // k bias
#pragma unroll
    for (int nb = 0; nb < 8; ++nb) {
        float bb = bk[nb * 16 + l15];
#pragma unroll
        for (int j = 0; j < 8; ++j) acc[nb][j] += bb;
    }
    // argmax over clusters (lanes 0..7 = clusters, first-index tie-break)
    int myassign = 0;                    // valid on lane l15==j for edge half*8+j
#pragma unroll
    for (int j = 0; j < 8; ++j) {
        float v = sacc[j];
        int idx = l15 & 7;
#pragma unroll
        for (int off = 1; off <= 4; off <<= 1) {
            float ov = __shfl_xor(v, off);
            int oi = __shfl_xor(idx, off);
            if (ov > v || (ov == v && oi < idx)) { v = ov; idx = oi; }
        }
        if (l15 == j) myassign = idx;
    }
    // attn = <q[src], k_row> per edge
#pragma unroll
    for (int j = 0; j < 8; ++j) {
        int jeff = half * 8 + j;
        int srcj = __shfl(mysrc, jeff);            // lane jeff (half0) holds it
        const float* qrow = q + (size_t)srcj * DD;
        float p = 0.0f;
#pragma unroll
        for (int nb = 0; nb < 8; ++nb) p += acc[nb][j] * qrow[nb * 16 + l15];
        p += __shfl_xor(p, 1);
        p += __shfl_xor(p, 2);
        p += __shfl_xor(p, 4);
        p += __shfl_xor(p, 8);                     // reduced within 16-lane half
        int e = eBase + jeff;
        if (l15 == j && e < NE) {
            float at = p * SCALE_V;
            attnOut[e] = at;
            int s = myassign * NN + srcj;
            segOut[e] = s;
            atomicAdd(&counts[s], 1);
            atomicMax(&segmax[s], ordEnc(at));
        }
    }
}

// ---------------------------------------------------------------- pass C: exp + denom
__global__ void passC_kernel(const float* __restrict__ attn, const int* __restrict__ seg,
                             const unsigned* __restrict__ segmax,
                             float* __restrict__ ex, float* __restrict__ denom) {
    int e = blockIdx.x * 256 + threadIdx.x;
    if (e >= NE) return;
    int s = seg[e];
    float m = ordDec(segmax[s]);
    float v = __expf(attn[e] - m);
    ex[e] = v;
    atomicAdd(&denom[s], v);
}

// ---------------------------------------------------------------- non-empty cluster flags
__global__ void flag_kernel(const int* __restrict__ counts, int* __restrict__ epcflag) {
    __shared__ int f[NCLUST];
    int tid = threadIdx.x;
    if (tid < NCLUST) f[tid] = 0;
    __syncthreads();
    int i = blockIdx.x * 256 + tid;
    if (i < CN && counts[i] > 0) f[i / NN] = 1;    // benign race
    __syncthreads();
    if (tid < NCLUST && f[tid]) atomicOr(&epcflag[tid], 1);
}

// ------------------------------------------- pass D: v GEMM + weighted scatter-add
__global__ __launch_bounds__(256) void passD_kernel(
    const float* __restrict__ TF, const float* __restrict__ Wv, const float* __restrict__ bvv,
    const float* __restrict__ ex, const float* __restrict__ denom, const int* __restrict__ seg,
    float* __restrict__ sums) {
    __shared__ float wlds[DD * DD];
    int tid = threadIdx.x;
    load_w_lds_swz(wlds, Wv, tid);
    __syncthreads();
    int wave = tid >> 5, lane = tid & 31;
    int half = lane >> 4, l15 = lane & 15, koff = half * 2;
    int cbase = l15 + (half << 4);
    int eBase = blockIdx.x * 128 + wave * 16;
    int ec = (eBase + l15 < NE) ? (eBase + l15) : 0;
    const float* trow = TF + (size_t)ec * DD;

    v8f acc[8] = {};
    v2f a = *(const v2f*)(trow + koff);                 // pipelined A load
    for (int kb = 0; kb < DD; kb += 4) {
        v2f an = *(const v2f*)(trow + ((kb + 4) & (DD - 1)) + koff);
#pragma unroll
        for (int nb = 0; nb < 8; ++nb) {
            int colp = (nb * 16 + cbase) & 127;
            acc[nb] = wmma_f32(a, wmma_b_ld(wlds, kb, half, colp), acc[nb]);
        }
        a = an;
    }
#pragma unroll
    for (int nb = 0; nb < 8; ++nb) {
        float bb = bvv[nb * 16 + l15];
#pragma unroll
        for (int j = 0; j < 8; ++j) acc[nb][j] += bb;
    }
    // softmax weight per edge (lane l15<8 of each half owns edge half*8+l15)
    float wloc = 0.0f;
    int sgloc = -1;
    if (l15 < 8) {
        int e = eBase + half * 8 + l15;
        if (e < NE) {
            int s = seg[e];
            wloc = ex[e] / denom[s];
            sgloc = s;
        }
    }
#pragma unroll
    for (int j = 0; j < 8; ++j) {
        int srcLane = (lane & 16) | j;
        float wj = __shfl(wloc, srcLane);
        int sgj = __shfl(sgloc, srcLane);
        if (sgj >= 0) {
            float* base = sums + (size_t)sgj * DD;
#pragma unroll
            for (int nb = 0; nb < 8; ++nb)
                atomicAdd(&base[nb * 16 + l15], wj * acc[nb][j]);
        }
    }
}

// ------------------------------------------- pass E: combine clusters + out GEMM + relu
__global__ __launch_bounds__(256) void passE_kernel(
    const float* __restrict__ sums, const int* __restrict__ counts,
    const int* __restrict__ epcflag, const float* __restrict__ Wo,
    const float* __restrict__ bo, float* __restrict__ out) {
    __shared__ float wlds[DD * DD];
    int tid = threadIdx.x;
    load_w_lds_swz(wlds, Wo, tid);
    __syncthreads();
    int wave = tid >> 5, lane = tid & 31;
    int half = lane >> 4, l15 = lane & 15, koff = half * 2;
    int cbase = l15 + (half << 4);
    int rowBase = blockIdx.x * 128 + wave * 16;
    int rc = (rowBase + l15 < NN) ? (rowBase + l15) : 0;

    int nne = 0;
#pragma unroll
    for (int c = 0; c < NCLUST; ++c) nne += epcflag[c];
    float inv_nn = (nne > 0) ? (1.0f / (float)nne) : 0.0f;

    float invc[NCLUST];
#pragma unroll
    for (int c = 0; c < NCLUST; ++c) {
        int cnt = counts[c * NN + rc];
        invc[c] = inv_nn / (float)(cnt > 0 ? cnt : 1);
    }
    const float* srow = sums + (size_t)rc * DD;

    v8f acc[8] = {};
    for (int kb = 0; kb < DD; kb += 4) {
        v2f a; a.x = 0.0f; a.y = 0.0f;
#pragma unroll
        for (int c = 0; c < NCLUST; ++c) {
            v2f sv = *(const v2f*)(srow + (size_t)c * (size_t)NN * DD + kb + koff);
            a.x += sv.x * invc[c];
            a.y += sv.y * invc[c];
        }
#pragma unroll
        for (int nb = 0; nb < 8; ++nb) {
            int colp = (nb * 16 + cbase) & 127;
            acc[nb] = wmma_f32(a, wmma_b_ld(wlds, kb, half, colp), acc[nb]);
        }
    }
#pragma unroll
    for (int nb = 0; nb < 8; ++nb) {
        int col = nb * 16 + l15;
        float bb = bo[col];
#pragma unroll
        for (int j = 0; j < 8; ++j) {
            int r = rowBase + half * 8 + j;
            if (r < NN) {
                float v = acc[nb][j] + bb;
                out[(size_t)r * DD + col] = v > 0.0f ? v : 0.0f;
            }
        }
    }
}

extern "C" void kernel_launch(void* const* d_in, const int* in_sizes, int n_in,
                              void* d_out, int out_size, void* d_ws, size_t ws_size,
                              hipStream_t stream) {
    const float* node_feat = (const float*)d_in[0];
    const float* time_feat = (const float*)d_in[1];
    /* d_in[2] context_feat: unused by reference */
    const int* edge_index  = (const int*)d_in[3];
    const float* Wq = (const float*)d_in[4];
    const float* bq = (const float*)d_in[5];
    const float* Wk = (const float*)d_in[6];
    const float* bk = (const float*)d_in[7];
    const float* Wv = (const float*)d_in[8];
    const float* bv = (const float*)d_in[9];
    const float* ce = (const float*)d_in[10];
    const float* Wo = (const float*)d_in[11];
    const float* bo = (const float*)d_in[12];
    float* out = (float*)d_out;

    char* ws = (char*)d_ws;                       // all offsets 256B-aligned
    float*    q      = (float*)(ws + 0);          // 25,600,000 B
    float*    attn   = (float*)(ws + 25600000);   //  2,400,000 B
    int*      seg    = (int*)(ws + 28000000);     //  2,400,000 B
    float*    ex     = (float*)(ws + 30400000);   //  2,400,000 B
    unsigned* smx    = (unsigned*)(ws + 32800000);//  1,600,000 B
    float*    denom  = (float*)(ws + 34400000);   //  1,600,000 B
    int*      counts = (int*)(ws + 36000000);     //  1,600,000 B
    int*      epcflg = (int*)(ws + 37600000);     //        256 B
    float*    sums   = (float*)(ws + 37600256);   // 204,800,000 B  (total ~242.4 MB)

    const int* srcIdx = edge_index;               // row 0 of [2,E]

    init_kernel<<<(int)((CND + 255) / 256), 256, 0, stream>>>(sums, denom, counts, smx, epcflg);
    qgemm_kernel<<<(NN + 127) / 128, 256, 0, stream>>>(node_feat, Wq, bq, q, NN);
    passB_kernel<<<(NE + 127) / 128, 256, 0, stream>>>(time_feat, Wk, bk, ce, q, srcIdx,
                                                       attn, seg, counts, smx);
    passC_kernel<<<(NE + 255) / 256, 256, 0, stream>>>(attn, seg, smx, ex, denom);
    flag_kernel<<<(CN + 255) / 256, 256, 0, stream>>>(counts, epcflg);
    passD_kernel<<<(NE + 127) / 128, 256, 0, stream>>>(time_feat, Wv, bv, ex, denom, seg, sums);
    passE_kernel<<<(NN + 127) / 128, 256, 0, stream>>>(sums, counts, epcflg, Wo, bo, out);
}